// GCN_2963527434974
// MI455X (gfx1250) — compile-verified
//
#include <hip/hip_runtime.h>
#include <hip/hip_bf16.h>

typedef __attribute__((ext_vector_type(2))) float v2f;
typedef __attribute__((ext_vector_type(8))) float v8f;

#define K_DIM 256

// ---------------- utility kernels ----------------

__global__ void zero_kernel(float* __restrict__ p, long n) {
  long i = (long)blockIdx.x * blockDim.x + threadIdx.x;
  if (i < n) p[i] = 0.0f;
}

__global__ void deg_kernel(const int* __restrict__ src, const int* __restrict__ dst,
                           float* __restrict__ ns, float* __restrict__ nd, int nE) {
  int e = blockIdx.x * blockDim.x + threadIdx.x;
  if (e < nE) {
    atomicAdd(&ns[src[e]], 1.0f);
    atomicAdd(&nd[dst[e]], 1.0f);
  }
}

__global__ void norm_kernel(float* __restrict__ ns, float* __restrict__ nd, int n) {
  int i = blockIdx.x * blockDim.x + threadIdx.x;
  if (i < n) {
    float a = ns[i]; ns[i] = (a > 0.0f) ? rsqrtf(a) : 1.0f;
    float b = nd[i]; nd[i] = (b > 0.0f) ? rsqrtf(b) : 1.0f;
  }
}

// Wt[n*K + k] = W[k*NF + n]   (makes B-fragment loads contiguous 8B)
template <int NF>
__global__ void transpose_kernel(const float* __restrict__ W, float* __restrict__ Wt) {
  int i = blockIdx.x * blockDim.x + threadIdx.x;
  if (i < K_DIM * NF) {
    int k = i / NF, n = i % NF;
    Wt[n * K_DIM + k] = W[i];
  }
}

// ---------------- WMMA f32 GEMM: H = diag(ns) * X * W ----------------
// One wave (32 threads) per 16x64 output strip.
// V_WMMA_F32_16X16X4_F32 operand layouts (ISA 7.12.2):
//   A 16x4 : lane l holds row (l&15); vgpr0/1 = K = 2*(l>>4), 2*(l>>4)+1
//   B 4x16 : lane l holds col (l&15); vgpr0/1 = K = 2*(l>>4), 2*(l>>4)+1
//   C 16x16: lane l holds col (l&15); vgpr r = row r + 8*(l>>4)
template <int NF>
__global__ __launch_bounds__(32)
void gemm_wmma_kernel(const float* __restrict__ X, const float* __restrict__ Wt,
                      const float* __restrict__ ns, float* __restrict__ H) {
  constexpr int STRIPS = NF / 64;
  const int tile = blockIdx.x;
  const int tn = tile % STRIPS;
  const int tm = tile / STRIPS;
  const int lane = threadIdx.x;
  const int half = lane >> 4;
  const int lid  = lane & 15;

  const int row = tm * 16 + lid;
  const float scale = ns[row];
  const float* ap  = X  + (long)row * K_DIM + 2 * half;
  const float* bp0 = Wt + (long)(tn * 64 + lid) * K_DIM + 2 * half;

  v8f c0 = {}, c1 = {}, c2 = {}, c3 = {};

#pragma unroll 8
  for (int s = 0; s < K_DIM / 4; ++s) {
    v2f a = *(const v2f*)(ap + 4 * s);
    a *= scale;  // fused norm_src row scaling
    v2f b0 = *(const v2f*)(bp0 + 4 * s);
    v2f b1 = *(const v2f*)(bp0 + 16 * K_DIM + 4 * s);
    v2f b2 = *(const v2f*)(bp0 + 32 * K_DIM + 4 * s);
    v2f b3 = *(const v2f*)(bp0 + 48 * K_DIM + 4 * s);
    c0 = __builtin_amdgcn_wmma_f32_16x16x4_f32(false, a, false, b0, (short)0, c0, false, false);
    c1 = __builtin_amdgcn_wmma_f32_16x16x4_f32(false, a, false, b1, (short)0, c1, false, false);
    c2 = __builtin_amdgcn_wmma_f32_16x16x4_f32(false, a, false, b2, (short)0, c2, false, false);
    c3 = __builtin_amdgcn_wmma_f32_16x16x4_f32(false, a, false, b3, (short)0, c3, false, false);
  }

  const int orow = tm * 16 + half * 8;
  float* out = H + (long)orow * NF + tn * 64 + lid;
#pragma unroll
  for (int r = 0; r < 8; ++r) {
    out[(long)r * NF +  0] = c0[r];
    out[(long)r * NF + 16] = c1[r];
    out[(long)r * NF + 32] = c2[r];
    out[(long)r * NF + 48] = c3[r];
  }
}

// ---------------- edge scatter: m[dst] += h[src] ----------------
// one thread per (edge, 4-float chunk); accumulator is L2-resident (51 MB << 192 MB L2)
template <int F>
__global__ void scatter_kernel(const float* __restrict__ H, const int* __restrict__ src,
                               const int* __restrict__ dst, float* __restrict__ Macc,
                               long total) {
  long i = (long)blockIdx.x * blockDim.x + threadIdx.x;
  if (i >= total) return;
  constexpr int CH = F / 4;
  const int e = (int)(i / CH);
  const int c = ((int)(i % CH)) << 2;
  const int s = src[e];
  const int d = dst[e];
  const float4 v = *(const float4*)(H + (long)s * F + c);
  float* mp = Macc + (long)d * F + c;
  atomicAdd(mp + 0, v.x);
  atomicAdd(mp + 1, v.y);
  atomicAdd(mp + 2, v.z);
  atomicAdd(mp + 3, v.w);
}

// ---------------- finalize: out = m * norm_dst + b (optional relu, dual write) ----------------
template <int F, bool RELU>
__global__ void finalize_kernel(const float* __restrict__ Macc, const float* __restrict__ nd,
                                const float* __restrict__ bias, float* __restrict__ out,
                                float* __restrict__ out2, long total) {
  long i = (long)blockIdx.x * blockDim.x + threadIdx.x;
  if (i >= total) return;
  const int node = (int)(i / F);
  const int f = (int)(i % F);
  float v = Macc[i] * nd[node] + bias[f];
  if (RELU) v = fmaxf(v, 0.0f);
  out[i] = v;
  if (out2) out2[i] = v;
}

// ---------------- host-side orchestration ----------------

extern "C" void kernel_launch(void* const* d_in, const int* in_sizes, int n_in,
                              void* d_out, int out_size, void* d_ws, size_t ws_size,
                              hipStream_t stream) {
  (void)in_sizes; (void)n_in; (void)out_size; (void)ws_size;

  const float* feats = (const float*)d_in[0];
  const int*   src   = (const int*)d_in[1];
  const int*   dst   = (const int*)d_in[2];
  const float* W0    = (const float*)d_in[3];
  const float* b0    = (const float*)d_in[4];
  const float* W1    = (const float*)d_in[5];
  const float* b1    = (const float*)d_in[6];
  const float* W2    = (const float*)d_in[7];
  const float* b2    = (const float*)d_in[8];

  const int N = 50000, E = 800000;

  // workspace layout (floats), 16-byte aligned offsets; total ~103 MB
  float* ws = (float*)d_ws;
  float* ns = ws;                         // 50000 (padded to 50048)
  float* nd = ws + 50048;                 // 50000
  float* Wt = ws + 100096;                // up to 65536
  float* hw = ws + 100096 + 65536;        // N*256 GEMM output
  float* mA = hw + (long)N * 256;         // N*256 aggregation accumulator

  float* out = (float*)d_out;
  float* h0  = out;                       // N*256
  float* h1  = out + (long)N * 256;       // N*256
  float* h2  = out + (long)N * 512;       // N*128
  float* h2b = out + (long)N * 640;       // N*128 (duplicate)

  const int T = 256;
  const long n256 = (long)N * 256;
  const long n128 = (long)N * 128;
  const long sc256 = (long)E * (256 / 4);
  const long sc128 = (long)E * (128 / 4);

  // degrees -> norms (graph-dependent, computed once)
  zero_kernel<<<(100096 + T - 1) / T, T, 0, stream>>>(ws, 100096);
  deg_kernel<<<(E + T - 1) / T, T, 0, stream>>>(src, dst, ns, nd, E);
  norm_kernel<<<(N + T - 1) / T, T, 0, stream>>>(ns, nd, N);

  // ---- layer 0: 256 -> 256, relu ----
  transpose_kernel<256><<<(256 * 256 + T - 1) / T, T, 0, stream>>>(W0, Wt);
  gemm_wmma_kernel<256><<<(N / 16) * (256 / 64), 32, 0, stream>>>(feats, Wt, ns, hw);
  zero_kernel<<<(int)((n256 + T - 1) / T), T, 0, stream>>>(mA, n256);
  scatter_kernel<256><<<(int)((sc256 + T - 1) / T), T, 0, stream>>>(hw, src, dst, mA, sc256);
  finalize_kernel<256, true><<<(int)((n256 + T - 1) / T), T, 0, stream>>>(mA, nd, b0, h0, nullptr, n256);

  // ---- layer 1: 256 -> 256, relu ----
  transpose_kernel<256><<<(256 * 256 + T - 1) / T, T, 0, stream>>>(W1, Wt);
  gemm_wmma_kernel<256><<<(N / 16) * (256 / 64), 32, 0, stream>>>(h0, Wt, ns, hw);
  zero_kernel<<<(int)((n256 + T - 1) / T), T, 0, stream>>>(mA, n256);
  scatter_kernel<256><<<(int)((sc256 + T - 1) / T), T, 0, stream>>>(hw, src, dst, mA, sc256);
  finalize_kernel<256, true><<<(int)((n256 + T - 1) / T), T, 0, stream>>>(mA, nd, b1, h1, nullptr, n256);

  // ---- layer 2: 256 -> 128, no relu, duplicated output ----
  transpose_kernel<128><<<(256 * 128 + T - 1) / T, T, 0, stream>>>(W2, Wt);
  gemm_wmma_kernel<128><<<(N / 16) * (128 / 64), 32, 0, stream>>>(h1, Wt, ns, hw);
  zero_kernel<<<(int)((n128 + T - 1) / T), T, 0, stream>>>(mA, n128);
  scatter_kernel<128><<<(int)((sc128 + T - 1) / T), T, 0, stream>>>(hw, src, dst, mA, sc128);
  finalize_kernel<128, false><<<(int)((n128 + T - 1) / T), T, 0, stream>>>(mA, nd, b2, h2, h2b, n128);
}